// RelationNetwork_57621281243635
// MI455X (gfx1250) — compile-verified
//
#include <hip/hip_runtime.h>

typedef __attribute__((ext_vector_type(2))) float v2f;
typedef __attribute__((ext_vector_type(8))) float v8f;

#define BATCH 4
#define NPTS  1024
#define DDIM  64
#define PAIRS 523776.0f   // N*(N-1)/2 for N=1024

// ---------------------------------------------------------------------------
// Kernel 1: weighted reductions over N into a 16x128 zero-padded tile.
//   ucat[b*128 + d]       = sum_i (N-1-i) * objects[b,i,d]   (d < 64)
//   ucat[b*128 + 64 + d]  = sum_i   i     * objects[b,i,d]
//   rows 4..15 are zeroed (WMMA M-padding) so kernel 2 needs no branches.
// Grid: 16 blocks x 256 threads (blocks >= BATCH only zero their row).
// ---------------------------------------------------------------------------
__global__ __launch_bounds__(256) void rn_reduce(const float* __restrict__ objects,
                                                 float* __restrict__ ucat) {
    const int b = blockIdx.x;
    const int t = threadIdx.x;

    if (b >= BATCH) {                     // padding rows: just zero 128 floats
        if (t < 128) ucat[b * 128 + t] = 0.0f;
        return;
    }

    const int d = t & 63;
    const int g = t >> 6;                 // 4 row-groups
    const float* base = objects + (size_t)b * NPTS * DDIM;

    float su = 0.f, sv = 0.f;
    for (int i = g; i < NPTS; i += 4) {
        float x = base[i * DDIM + d];
        su += (float)(NPTS - 1 - i) * x;
        sv += (float)i * x;
    }

    __shared__ float lds[2][4][DDIM];
    lds[0][g][d] = su;
    lds[1][g][d] = sv;
    __syncthreads();

    if (g == 0) {
        float tu = lds[0][0][d] + lds[0][1][d] + lds[0][2][d] + lds[0][3][d];
        float tv = lds[1][0][d] + lds[1][1][d] + lds[1][2][d] + lds[1][3][d];
        ucat[b * 128 + d]      = tu;
        ucat[b * 128 + 64 + d] = tv;
    }
}

// ---------------------------------------------------------------------------
// Kernel 2: s = Ucat(16x128, pre-padded) @ Wg^T(128x64) + P*bg
//           out = s(16x64) @ Wf^T(64x64) + bf            (rows 0..3 stored)
// One block, 128 threads = 4 wave32s; wave w owns 16-column tile e0/f0 = 16w.
// V_WMMA_F32_16X16X4_F32 chains; all fragment loads are branch-free b64.
// ---------------------------------------------------------------------------
__global__ __launch_bounds__(128) void rn_gemm(const float* __restrict__ ucat,
                                               const float* __restrict__ Wg,
                                               const float* __restrict__ bg,
                                               const float* __restrict__ Wf,
                                               const float* __restrict__ bf,
                                               float* __restrict__ out) {
    const int t    = threadIdx.x;
    const int lane = t & 31;
    const int w    = t >> 5;        // wave id 0..3
    const int half = lane >> 4;     // lane half selects K offset (+2) per ISA layout
    const int l    = lane & 15;
    const int koff = half * 2;
    const int e0   = w * 16;

    __shared__ float s_lds[16 * 64];

    // Per-lane base pointers (all 8-byte aligned: k multiple of 4, koff even)
    const v2f* __restrict__ pa = (const v2f*)(ucat + l * 128 + koff);            // A: Ucat[M=l, K]
    const v2f* __restrict__ pg = (const v2f*)(Wg + (e0 + l) * 128 + koff);       // B: Wg[e0+l, K]
    const v2f* __restrict__ pf = (const v2f*)(Wf + (e0 + l) * 64 + koff);        // B: Wf[f0+l, K]

    // ---- Stage A: s-tile (columns e0..e0+15), K = 128, 32 WMMAs ----
    v8f c = {};
#pragma unroll
    for (int k = 0; k < 32; ++k) {
        v2f a  = pa[k * 2];   // Ucat[l, 4k + koff + {0,1}]
        v2f bm = pg[k * 2];   // Wg[e0+l, 4k + koff + {0,1}]
        c = __builtin_amdgcn_wmma_f32_16x16x4_f32(false, a, false, bm,
                                                  (short)0, c, false, false);
    }
    // + P * bg[e] (same column e = e0+l for every C register of this lane)
    const float pbias = PAIRS * bg[e0 + l];
#pragma unroll
    for (int r = 0; r < 8; ++r) c[r] += pbias;

    // C layout: vgpr r -> row M = r + 8*half, col N = e0 + l
#pragma unroll
    for (int r = 0; r < 8; ++r) s_lds[(r + half * 8) * 64 + (e0 + l)] = c[r];
    __syncthreads();

    // ---- Stage B: out-tile (columns f0..f0+15), K = 64, 16 WMMAs ----
    const int f0 = w * 16;
    const v2f* __restrict__ ps = (const v2f*)(s_lds + l * 64 + koff);            // A: s[M=l, K]
    v8f c2 = {};
#pragma unroll
    for (int k = 0; k < 16; ++k) {
        v2f a  = ps[k * 2];   // s[l, 4k + koff + {0,1}]
        v2f bm = pf[k * 2];   // Wf[f0+l, 4k + koff + {0,1}]
        c2 = __builtin_amdgcn_wmma_f32_16x16x4_f32(false, a, false, bm,
                                                   (short)0, c2, false, false);
    }
    const float bias2 = bf[f0 + l];
#pragma unroll
    for (int r = 0; r < 8; ++r) c2[r] += bias2;

    // Rows 0..3 live in vgprs 0..3 of the low lane-half. out is (4,64) f32.
    if (half == 0) {
#pragma unroll
        for (int r = 0; r < BATCH; ++r) out[r * 64 + (f0 + l)] = c2[r];
    }
}

// ---------------------------------------------------------------------------
extern "C" void kernel_launch(void* const* d_in, const int* in_sizes, int n_in,
                              void* d_out, int out_size, void* d_ws, size_t ws_size,
                              hipStream_t stream) {
    const float* objects = (const float*)d_in[0]; // (4,1024,64)
    const float* Wg      = (const float*)d_in[1]; // (64,128)
    const float* bg      = (const float*)d_in[2]; // (64,)
    const float* Wf      = (const float*)d_in[3]; // (64,64)
    const float* bf      = (const float*)d_in[4]; // (64,)
    float*       out     = (float*)d_out;         // (4,64)
    float*       ucat    = (float*)d_ws;          // 16*128 floats scratch (padded)

    rn_reduce<<<16, 256, 0, stream>>>(objects, ucat);
    rn_gemm<<<1, 128, 0, stream>>>(ucat, Wg, bg, Wf, bf, out);
}